// GNNImportancePredictor_91036126806851
// MI455X (gfx1250) — compile-verified
//
#include <hip/hip_runtime.h>
#include <hip/hip_bf16.h>
#include <math.h>

// ---------------------------------------------------------------------------
// GCN (3x GCNConv + sigmoid head), fp32 throughout to match the reference.
// GEMMs: V_WMMA_F32_16X16X4_F32, one wave = 16x64 output tile, B staged in
// LDS (transposed + padded -> ds_load_b64, no bank conflicts), K templated
// so the K loop fully unrolls with branch-free loads.
// Edge aggregation is L2-resident (3x 25.6MB buffers << 192MB L2), float4
// gathers + global_atomic_add_f32 scatters.
// ---------------------------------------------------------------------------

typedef __attribute__((ext_vector_type(2))) float v2f;
typedef __attribute__((ext_vector_type(8))) float v8f;

#define HID 64

// ---------------- degree / normalization ----------------
__global__ void deg_init_kernel(float* __restrict__ deg, int n) {
    int i = blockIdx.x * blockDim.x + threadIdx.x;
    if (i < n) deg[i] = 1.0f;   // self-loop
}

__global__ void deg_scatter_kernel(const int* __restrict__ dst,
                                   float* __restrict__ deg, int nE) {
    int e = blockIdx.x * blockDim.x + threadIdx.x;
    if (e < nE) atomicAdd(&deg[dst[e]], 1.0f);
}

__global__ void deg_rsqrt_kernel(float* __restrict__ deg, int n) {
    int i = blockIdx.x * blockDim.x + threadIdx.x;
    if (i < n) deg[i] = rsqrtf(deg[i]);
}

// ---------------- WMMA GEMM: Ht[N,64] = Hin[N,K] @ W[K,64] ----------------
// K is compile-time (3 or 64). W is staged in LDS transposed:
//   Wlds[col*WSTR + k] = W[k*64 + col], rows k in [K, KP) zero-padded.
// One wave computes a 16x64 tile: 4 accumulators, KP/4 fully unrolled ksteps.
template <int K>
__global__ __launch_bounds__(256) void gemm_wmma_f32_kernel(
    const float* __restrict__ Hin,
    const float* __restrict__ W,       // K x 64 row-major
    float* __restrict__ Ht, int nRows) {
    constexpr int KP   = (K + 3) & ~3;     // K padded to multiple of 4
    constexpr int WSTR = KP + 2;           // pad stride: kill LDS bank conflicts
    __shared__ float Wlds[HID * WSTR];

    // Cooperative transpose-load of W into LDS (branch-free for K%4 != 0:
    // clamped address + value select -> v_cndmask, unconditional loads).
    for (int idx = threadIdx.x; idx < HID * KP; idx += 256) {
        const int col = idx / KP;
        const int k   = idx % KP;
        if constexpr (K % 4 == 0) {
            Wlds[col * WSTR + k] = W[k * HID + col];
        } else {
            const int kc  = (k < K) ? k : (K - 1);
            const float v = W[kc * HID + col];
            Wlds[col * WSTR + k] = (k < K) ? v : 0.0f;
        }
    }
    __syncthreads();

    const int lane    = threadIdx.x & 31;
    const int waveId  = threadIdx.x >> 5;
    const int tile    = blockIdx.x * 8 + waveId;   // 16-row tile index
    const int row0    = tile * 16;
    if (row0 >= nRows) return;                     // wave-uniform; EXEC stays full
    const int halfSel = lane >> 4;                 // 0: lanes 0-15, 1: lanes 16-31
    const int rlane   = lane & 15;
    const int rowg    = row0 + rlane;              // A-matrix row for this lane

    v8f c0 = {}, c1 = {}, c2 = {}, c3 = {};
    const float* __restrict__ arow = Hin + (size_t)rowg * K;
    const float* __restrict__ b0p  = &Wlds[(0 * 16 + rlane) * WSTR];
    const float* __restrict__ b1p  = &Wlds[(1 * 16 + rlane) * WSTR];
    const float* __restrict__ b2p  = &Wlds[(2 * 16 + rlane) * WSTR];
    const float* __restrict__ b3p  = &Wlds[(3 * 16 + rlane) * WSTR];

    #pragma unroll
    for (int ks = 0; ks < KP / 4; ++ks) {
        const int k0 = ks * 4 + halfSel * 2;       // f32 A/B layout: K split by half-wave
        v2f a;
        if constexpr (K % 4 == 0) {
            a = *(const v2f*)(arow + k0);          // one global_load_b64
        } else {
            // clamped addresses (always in bounds) + value cndmask, no branches
            const int i0 = (k0     < K) ? k0       : (K - 1);
            const int i1 = (k0 + 1 < K) ? (k0 + 1) : (K - 1);
            const float a0 = arow[i0];
            const float a1 = arow[i1];
            a.x = (k0     < K) ? a0 : 0.0f;
            a.y = (k0 + 1 < K) ? a1 : 0.0f;
        }
        const v2f b0 = *(const v2f*)(b0p + k0);    // ds_load_b64 each
        const v2f b1 = *(const v2f*)(b1p + k0);
        const v2f b2 = *(const v2f*)(b2p + k0);
        const v2f b3 = *(const v2f*)(b3p + k0);
        c0 = __builtin_amdgcn_wmma_f32_16x16x4_f32(false, a, false, b0, (short)0, c0, false, false);
        c1 = __builtin_amdgcn_wmma_f32_16x16x4_f32(false, a, false, b1, (short)0, c1, false, false);
        c2 = __builtin_amdgcn_wmma_f32_16x16x4_f32(false, a, false, b2, (short)0, c2, false, false);
        c3 = __builtin_amdgcn_wmma_f32_16x16x4_f32(false, a, false, b3, (short)0, c3, false, false);
    }

    // D layout: VGPR r -> M = r + halfSel*8, N = rlane (within each 16-col tile)
    #pragma unroll
    for (int r = 0; r < 8; ++r) {
        const int rowl = r + halfSel * 8;
        float* __restrict__ orow = Ht + (size_t)(row0 + rowl) * HID + rlane;
        orow[0]  = c0[r];
        orow[16] = c1[r];
        orow[32] = c2[r];
        orow[48] = c3[r];
    }
}

// ---------------- zero the aggregation buffer ----------------
__global__ void zero_kernel(float* __restrict__ p, int n) {
    int i = blockIdx.x * blockDim.x + threadIdx.x;
    if (i < n) p[i] = 0.0f;
}

// ---------------- edge scatter: Agg[d] += Ht[s] * dinv[s]*dinv[d] ----------
// 16 threads per edge, 4 floats per thread (float4 gather + 4 f32 atomics).
__global__ void edge_scatter_kernel(const int* __restrict__ src,
                                    const int* __restrict__ dst,
                                    const float* __restrict__ dinv,
                                    const float* __restrict__ Ht,
                                    float* __restrict__ Agg, int nE) {
    int gid = blockIdx.x * blockDim.x + threadIdx.x;
    int e = gid >> 4;
    if (e >= nE) return;
    int j = (gid & 15) * 4;
    int s = src[e];
    int d = dst[e];
    float w = dinv[s] * dinv[d];
    const float4 v = *(const float4*)(Ht + (size_t)s * HID + j);
    float* ap = Agg + (size_t)d * HID + j;
    atomicAdd(ap + 0, v.x * w);
    atomicAdd(ap + 1, v.y * w);
    atomicAdd(ap + 2, v.z * w);
    atomicAdd(ap + 3, v.w * w);
}

// ---------------- finalize: H = relu(Agg + Ht*dinv^2 + b) ------------------
__global__ void finalize_kernel(const float* __restrict__ Ht,
                                const float* __restrict__ dinv,
                                const float* __restrict__ bias,
                                float* __restrict__ Agg,  // in-out -> h_next
                                int nNodes) {
    int gid = blockIdx.x * blockDim.x + threadIdx.x;
    int i = gid >> 4;
    if (i >= nNodes) return;
    int j = (gid & 15) * 4;
    float w = dinv[i] * dinv[i];
    const float4 hv = *(const float4*)(Ht + (size_t)i * HID + j);
    float4 av = *(float4*)(Agg + (size_t)i * HID + j);
    av.x = fmaxf(av.x + hv.x * w + bias[j + 0], 0.0f);
    av.y = fmaxf(av.y + hv.y * w + bias[j + 1], 0.0f);
    av.z = fmaxf(av.z + hv.z * w + bias[j + 2], 0.0f);
    av.w = fmaxf(av.w + hv.w * w + bias[j + 3], 0.0f);
    *(float4*)(Agg + (size_t)i * HID + j) = av;
}

// ---------------- head: out = sigmoid(H @ Wo + bo) -------------------------
__global__ void head_kernel(const float* __restrict__ H,
                            const float* __restrict__ Wo,
                            const float* __restrict__ bo,
                            float* __restrict__ out, int nNodes) {
    int i = blockIdx.x * blockDim.x + threadIdx.x;
    if (i >= nNodes) return;
    const float* hp = H + (size_t)i * HID;
    float acc = 0.0f;
    #pragma unroll
    for (int j = 0; j < HID; j += 4) {
        float4 hv = *(const float4*)(hp + j);
        float4 wv = *(const float4*)(Wo + j);
        acc += hv.x * wv.x + hv.y * wv.y + hv.z * wv.z + hv.w * wv.w;
    }
    acc += bo[0];
    out[i] = 1.0f / (1.0f + __expf(-acc));
}

static inline int cdiv(long long a, long long b) { return (int)((a + b - 1) / b); }

extern "C" void kernel_launch(void* const* d_in, const int* in_sizes, int n_in,
                              void* d_out, int out_size, void* d_ws, size_t ws_size,
                              hipStream_t stream) {
    const float* x   = (const float*)d_in[0];   // [N,3]
    const int*   ei  = (const int*)  d_in[1];   // [2,E]
    const float* W0  = (const float*)d_in[2];   // [3,64]
    const float* b0  = (const float*)d_in[3];
    const float* W1  = (const float*)d_in[4];   // [64,64]
    const float* b1  = (const float*)d_in[5];
    const float* W2  = (const float*)d_in[6];   // [64,64]
    const float* b2  = (const float*)d_in[7];
    const float* Wo  = (const float*)d_in[8];   // [64,1]
    const float* bo  = (const float*)d_in[9];

    const int N = in_sizes[0] / 3;
    const int E = in_sizes[1] / 2;
    const int* src = ei;
    const int* dst = ei + E;

    float* ws   = (float*)d_ws;
    float* dinv = ws;                       // [N]      (deg, then rsqrt in place)
    float* buf0 = dinv + N;                 // [N,64]   Ht scratch
    float* buf1 = buf0 + (size_t)N * HID;   // [N,64]
    float* buf2 = buf1 + (size_t)N * HID;   // [N,64]

    const int B = 256;
    const int gemmBlocks = cdiv(cdiv(N, 16), 8);         // 8 waves (16-row tiles) / block
    const int featThreads = N * 16;                      // (node, 4 floats)
    const long long edgeThreads = (long long)E * 16;     // (edge, 4 floats)

    // --- normalization: deg = 1 + indegree; dinv = rsqrt(deg)
    deg_init_kernel<<<cdiv(N, B), B, 0, stream>>>(dinv, N);
    deg_scatter_kernel<<<cdiv(E, B), B, 0, stream>>>(dst, dinv, E);
    deg_rsqrt_kernel<<<cdiv(N, B), B, 0, stream>>>(dinv, N);

    // --- layer 0: in = x (K=3), ht = buf0, agg/out = buf1
    gemm_wmma_f32_kernel<3><<<gemmBlocks, B, 0, stream>>>(x, W0, buf0, N);
    zero_kernel<<<cdiv((long long)N * HID, B), B, 0, stream>>>(buf1, N * HID);
    edge_scatter_kernel<<<cdiv(edgeThreads, B), B, 0, stream>>>(src, dst, dinv, buf0, buf1, E);
    finalize_kernel<<<cdiv(featThreads, B), B, 0, stream>>>(buf0, dinv, b0, buf1, N);

    // --- layer 1: in = buf1, ht = buf0, agg/out = buf2
    gemm_wmma_f32_kernel<64><<<gemmBlocks, B, 0, stream>>>(buf1, W1, buf0, N);
    zero_kernel<<<cdiv((long long)N * HID, B), B, 0, stream>>>(buf2, N * HID);
    edge_scatter_kernel<<<cdiv(edgeThreads, B), B, 0, stream>>>(src, dst, dinv, buf0, buf2, E);
    finalize_kernel<<<cdiv(featThreads, B), B, 0, stream>>>(buf0, dinv, b1, buf2, N);

    // --- layer 2: in = buf2, ht = buf0, agg/out = buf1
    gemm_wmma_f32_kernel<64><<<gemmBlocks, B, 0, stream>>>(buf2, W2, buf0, N);
    zero_kernel<<<cdiv((long long)N * HID, B), B, 0, stream>>>(buf1, N * HID);
    edge_scatter_kernel<<<cdiv(edgeThreads, B), B, 0, stream>>>(src, dst, dinv, buf0, buf1, E);
    finalize_kernel<<<cdiv(featThreads, B), B, 0, stream>>>(buf0, dinv, b2, buf1, N);

    // --- head
    head_kernel<<<cdiv(N, B), B, 0, stream>>>(buf1, Wo, bo, (float*)d_out, N);

    (void)n_in; (void)out_size; (void)ws_size;
}